// LuongAttention_13116830122555
// MI455X (gfx1250) — compile-verified
//
#include <hip/hip_runtime.h>

#define B_ 32
#define T_ 4096
#define H_ 1024

typedef __attribute__((ext_vector_type(2))) float v2f;
typedef __attribute__((ext_vector_type(8))) float v8f;

// ---------------------------------------------------------------------------
// Pass 1: scores[t] = sum_{b,h} dec[b,h] * enc[b,t,h]
// One workgroup per t. Coalesced float4 loads along h; dec stays hot in L2.
// ---------------------------------------------------------------------------
__global__ __launch_bounds__(256) void luong_scores_kernel(
    const float* __restrict__ dec,
    const float* __restrict__ enc,
    float* __restrict__ scores)
{
    const int t   = blockIdx.x;
    const int tid = threadIdx.x;
    const int h0  = tid * 4;

    float acc = 0.0f;
#pragma unroll 4
    for (int b = 0; b < B_; ++b) {
        const float4 d = *(const float4*)(dec + (size_t)b * H_ + h0);
        const float4 e = *(const float4*)(enc + ((size_t)b * T_ + t) * H_ + h0);
        acc += d.x * e.x + d.y * e.y + d.z * e.z + d.w * e.w;
    }

    __shared__ float red[256];
    red[tid] = acc;
    __syncthreads();
#pragma unroll
    for (int s = 128; s > 0; s >>= 1) {
        if (tid < s) red[tid] += red[tid + s];
        __syncthreads();
    }
    if (tid == 0) scores[t] = red[0];
}

// ---------------------------------------------------------------------------
// Pass 2: dist = softmax(scores) over T elements. Single workgroup.
// ---------------------------------------------------------------------------
__global__ __launch_bounds__(256) void luong_softmax_kernel(
    const float* __restrict__ scores,
    float* __restrict__ dist)
{
    const int tid = threadIdx.x;
    __shared__ float red[256];

    // max-reduce
    float m = -3.402823466e38f;
    for (int i = tid; i < T_; i += 256) m = fmaxf(m, scores[i]);
    red[tid] = m;
    __syncthreads();
#pragma unroll
    for (int s = 128; s > 0; s >>= 1) {
        if (tid < s) red[tid] = fmaxf(red[tid], red[tid + s]);
        __syncthreads();
    }
    m = red[0];
    __syncthreads();

    // sum-exp reduce
    float sum = 0.0f;
    for (int i = tid; i < T_; i += 256) sum += __expf(scores[i] - m);
    red[tid] = sum;
    __syncthreads();
#pragma unroll
    for (int s = 128; s > 0; s >>= 1) {
        if (tid < s) red[tid] += red[tid + s];
        __syncthreads();
    }
    const float inv = 1.0f / red[0];

    for (int i = tid; i < T_; i += 256) dist[i] = __expf(scores[i] - m) * inv;
}

// ---------------------------------------------------------------------------
// Pass 3: context[b,h] = sum_t dist[t] * enc[b,t,h] via V_WMMA_F32_16X16X4_F32.
// A(16x4) = dist[t0..t0+3] broadcast across all 16 rows -> every row of D is
// the same context partial for the 16-wide h tile. Accumulate over t in C.
// Grid: (b, h-tile) = 32 x 64 workgroups; 8 waves split the T dimension.
// ---------------------------------------------------------------------------
__global__ __launch_bounds__(256) void luong_context_kernel(
    const float* __restrict__ enc,
    const float* __restrict__ dist,
    float* __restrict__ out)
{
    const int bid = blockIdx.x;
    const int b   = bid >> 6;          // batch index 0..31
    const int hb  = (bid & 63) << 4;   // h-tile base: 0,16,...,1008

    const int tid  = threadIdx.x;
    const int lane = tid & 31;         // wave32
    const int w    = tid >> 5;         // wave id 0..7
    const int lhi  = lane >> 4;        // 0 for lanes 0-15, 1 for lanes 16-31
    const int llo  = lane & 15;

    const float* __restrict__ encB = enc + (size_t)b * T_ * H_ + hb;

    v8f acc0 = {}; v8f acc1 = {}; v8f acc2 = {}; v8f acc3 = {};

    // chunk = w + 8*k, k = 0..127; t0 = 4*chunk covers all T=4096.
#define WMMA_STEP(ACC, CHUNK)                                                  \
    do {                                                                       \
        const int t0_ = 4 * (CHUNK);                                           \
        /* A-matrix (16x4 f32): VGPR0 = K0 (lanes 0-15) / K2 (16-31),          \
           VGPR1 = K1 / K3  -> float2 load of dist[t0 + 2*lhi]. */             \
        v2f Av = *(const v2f*)(dist + t0_ + 2 * lhi);                          \
        /* B-matrix (4x16 f32): VGPR0 = row K0/K1, VGPR1 = row K2/K3. */       \
        const float* p_ = encB + (size_t)(t0_ + lhi) * H_ + llo;               \
        v2f Bv;                                                                \
        Bv.x = p_[0];                                                          \
        Bv.y = p_[2 * H_];                                                     \
        ACC = __builtin_amdgcn_wmma_f32_16x16x4_f32(                           \
            false, Av, false, Bv, (short)0, ACC, false, false);                \
    } while (0)

#pragma unroll 1
    for (int j = 0; j < 32; ++j) {
        const int cbase = 32 * j + w;
        WMMA_STEP(acc0, cbase + 0);
        WMMA_STEP(acc1, cbase + 8);
        WMMA_STEP(acc2, cbase + 16);
        WMMA_STEP(acc3, cbase + 24);
    }
#undef WMMA_STEP

    acc0 += acc1;
    acc2 += acc3;
    acc0 += acc2;

    // D row M=0 lives in VGPR0 of lanes 0-15 (N = lane).
    const float r = acc0[0];

    __shared__ float red[8 * 16];
    if (lane < 16) red[w * 16 + lane] = r;
    __syncthreads();

    if (tid < 16) {
        float s = 0.0f;
#pragma unroll
        for (int ww = 0; ww < 8; ++ww) s += red[ww * 16 + tid];
        out[(size_t)b * H_ + hb + tid] = s;
    }
}

// ---------------------------------------------------------------------------
extern "C" void kernel_launch(void* const* d_in, const int* in_sizes, int n_in,
                              void* d_out, int out_size, void* d_ws, size_t ws_size,
                              hipStream_t stream) {
    (void)in_sizes; (void)n_in; (void)out_size; (void)ws_size;

    const float* dec = (const float*)d_in[0];   // [B, H]
    const float* enc = (const float*)d_in[1];   // [B, T, H]
    float* out    = (float*)d_out;              // [B, H]
    float* scores = (float*)d_ws;               // T floats
    float* dist   = scores + T_;                // T floats

    luong_scores_kernel <<<T_, 256, 0, stream>>>(dec, enc, scores);
    luong_softmax_kernel<<<1, 256, 0, stream>>>(scores, dist);
    luong_context_kernel<<<B_ * (H_ / 16), 256, 0, stream>>>(enc, dist, out);
}